// SWGCN_26439818674720
// MI455X (gfx1250) — compile-verified
//
#include <hip/hip_runtime.h>
#include <math.h>

#define NU 100000
#define NI 50000
#define NN 150000
#define DD 64
#define NB 3
#define EE 1000000
#define NHEAD 4
#define NBATCH 4096

// workspace offsets (in floats)
#define OFF_MAX  0ull                 // 3*NU uint keys
#define OFF_SUM  300000ull            // 3*NU f32
#define OFF_AGG  600000ull            // NI*3*64
#define OFF_ACT  10200000ull          // 3*E
#define OFF_Q    13200000ull          // 3*NN*64
#define OFF_K    42000000ull
#define OFF_V    70800000ull
#define OFF_ATT  99600000ull          // 3*NN*64
#define OFF_G    128400000ull         // NN*64

// d_out offsets (floats): emb(12288*64) | tpw(3*E) | aux(3*E)
#define OUT_TPW 786432ull
#define OUT_AUX 3786432ull

typedef float v2f __attribute__((ext_vector_type(2)));
typedef float v8f __attribute__((ext_vector_type(8)));

__device__ __forceinline__ unsigned fkey(float f) {
  unsigned b = __float_as_uint(f);
  return (b & 0x80000000u) ? ~b : (b | 0x80000000u);
}
__device__ __forceinline__ float fkey_inv(unsigned k) {
  unsigned b = (k & 0x80000000u) ? (k & 0x7fffffffu) : ~k;
  return __uint_as_float(b);
}

__global__ void k_zero(float* ws, unsigned n) {
  unsigned i = blockIdx.x * blockDim.x + threadIdx.x;
  if (i < n) ws[i] = 0.0f;
}

// ---- per-edge: act (pre-softmax), aux, atomic segment-max ----
__global__ void k_edge_phase(const float* __restrict__ user_emb,
                             const float* __restrict__ item_emb,
                             const float* __restrict__ alpha_w,
                             const float* __restrict__ alpha_b,
                             const int* __restrict__ ei,
                             float* __restrict__ ws, float* __restrict__ out) {
  int lane = threadIdx.x & 31;
  long long eid = (long long)blockIdx.x * 8 + (threadIdx.x >> 5);
  if (eid >= (long long)NB * EE) return;
  int b = (int)(eid / EE), e = (int)(eid % EE);
  int u = ei[(size_t)b * 2 * EE + e];
  int it = ei[(size_t)b * 2 * EE + EE + e];
  const float2* ue = (const float2*)(user_emb + ((size_t)u * NB + b) * DD);
  const float2* ie = (const float2*)(item_emb + ((size_t)it * NB + b) * DD);
  float2 a = ue[lane], c = ie[lane];
  float dx = fabsf(a.x - c.x), dy = fabsf(a.y - c.y);
  float2 aw = ((const float2*)(alpha_w + (size_t)b * DD))[lane];
  float pd = dx * aw.x + dy * aw.y;
  float pq = dx * dx + dy * dy;
  #pragma unroll
  for (int m = 16; m >= 1; m >>= 1) {
    pd += __shfl_xor(pd, m, 32);
    pq += __shfl_xor(pq, m, 32);
  }
  if (lane == 0) {
    float t = pd + alpha_b[b];
    float act = t > 0.f ? t : (expf(t) - 1.f);   // elu
    ws[OFF_ACT + (size_t)b * EE + e] = act;
    out[OUT_AUX + (size_t)b * EE + e] = pq;
    atomicMax((unsigned*)(ws + OFF_MAX) + (size_t)b * NU + u, fkey(act));
  }
}

// ---- per-edge: e = exp(act-max), atomic segment-sum ----
__global__ void k_edge_exp(const int* __restrict__ ei,
                           float* __restrict__ ws, float* __restrict__ out) {
  long long eid = (long long)blockIdx.x * blockDim.x + threadIdx.x;
  if (eid >= (long long)NB * EE) return;
  int b = (int)(eid / EE), e = (int)(eid % EE);
  int u = ei[(size_t)b * 2 * EE + e];
  float m = fkey_inv(((unsigned*)(ws + OFF_MAX))[(size_t)b * NU + u]);
  float ex = expf(ws[OFF_ACT + (size_t)b * EE + e] - m);
  atomicAdd(ws + OFF_SUM + (size_t)b * NU + u, ex);
  out[OUT_TPW + (size_t)b * EE + e] = ex;
}

// ---- per-edge: normalize tpw; scatter tpw*user_emb into agg ----
__global__ void k_norm_scatter(const float* __restrict__ user_emb,
                               const int* __restrict__ ei,
                               float* __restrict__ ws, float* __restrict__ out) {
  int lane = threadIdx.x & 31;
  long long eid = (long long)blockIdx.x * 8 + (threadIdx.x >> 5);
  if (eid >= (long long)NB * EE) return;
  int b = (int)(eid / EE), e = (int)(eid % EE);
  int u = ei[(size_t)b * 2 * EE + e];
  int it = ei[(size_t)b * 2 * EE + EE + e];
  float w = out[OUT_TPW + (size_t)b * EE + e] / ws[OFF_SUM + (size_t)b * NU + u];
  if (lane == 0) out[OUT_TPW + (size_t)b * EE + e] = w;
  const float2* ue = (const float2*)(user_emb + ((size_t)u * NB + b) * DD);
  float2 s = ue[lane];
  float* dst = ws + OFF_AGG + ((size_t)it * NB + b) * DD + 2 * lane;
  atomicAdd(dst, w * s.x);
  atomicAdd(dst + 1, w * s.y);
}

// ---- Q/K/V projections via V_WMMA_F32_16X16X4_F32; wave = 16-node tile ----
// LDS layouts tuned so both WMMA operands load as single aligned ds_load_b64:
//  - W pair-interleaved: (W[k][n], W[k+1][n]) contiguous for even k
//  - X tile rows padded to 68 floats (banks spread), fragments 8B-aligned
__global__ void __launch_bounds__(64)
k_qkv(const float* __restrict__ user_emb, const float* __restrict__ item_emb,
      const float* __restrict__ wq, const float* __restrict__ bq,
      const float* __restrict__ wk, const float* __restrict__ bk,
      const float* __restrict__ wv, const float* __restrict__ bv,
      float* __restrict__ ws) {
  __shared__ __align__(16) float sWp[DD * DD];
  __shared__ __align__(16) float sB[DD];
  __shared__ __align__(16) float sX[2][16 * 68];
  int wid = threadIdx.x >> 5, lane = threadIdx.x & 31;
  int half = lane >> 4, m16 = lane & 15;
  long long t = (long long)blockIdx.x * 2 + wid;
  long long tmax = (long long)NB * (NN / 16);
  if (t >= tmax) t = tmax - 1;       // duplicate waves store identical data
  int b = (int)(t / (NN / 16));
  int node0 = (int)(t % (NN / 16)) * 16;

  float* xt = sX[wid];
  for (int idx = lane; idx < 256; idx += 32) {
    int row = idx >> 4, seg = idx & 15, node = node0 + row;
    float4 xv;
    if (node < NU) {
      xv = *(const float4*)(user_emb + ((size_t)node * NB + b) * DD + seg * 4);
    } else {
      size_t o = ((size_t)(node - NU) * NB + b) * DD + seg * 4;
      float4 iv = *(const float4*)(item_emb + o);
      float4 av = *(const float4*)(ws + OFF_AGG + o);
      xv.x = iv.x + 2.f * av.x; xv.y = iv.y + 2.f * av.y;
      xv.z = iv.z + 2.f * av.z; xv.w = iv.w + 2.f * av.w;
    }
    xt[row * 68 + seg * 4 + 0] = xv.x; xt[row * 68 + seg * 4 + 1] = xv.y;
    xt[row * 68 + seg * 4 + 2] = xv.z; xt[row * 68 + seg * 4 + 3] = xv.w;
  }

  const float* Wm[3] = {wq, wk, wv};
  const float* Bm[3] = {bq, bk, bv};
  const size_t Qo[3] = {OFF_Q, OFF_K, OFF_V};
  #pragma unroll
  for (int w = 0; w < 3; ++w) {
    __syncthreads();
    for (int i = threadIdx.x; i < DD * DD; i += 64) {
      int k = i >> 6, n = i & 63;
      sWp[(k >> 1) * 128 + n * 2 + (k & 1)] = Wm[w][i];
    }
    for (int i = threadIdx.x; i < DD; i += 64) sB[i] = Bm[w][i];
    __syncthreads();
    for (int ct = 0; ct < 4; ++ct) {
      float bias = sB[ct * 16 + m16];
      v8f acc = {bias, bias, bias, bias, bias, bias, bias, bias};
      #pragma unroll
      for (int ks = 0; ks < 16; ++ks) {
        int kb = ks * 4 + 2 * half;    // A half-lanes hold K {0,1}/{2,3}
        v2f a = *(const v2f*)(xt + m16 * 68 + kb);
        v2f bb = *(const v2f*)(sWp + (kb >> 1) * 128 + (ct * 16 + m16) * 2);
        acc = __builtin_amdgcn_wmma_f32_16x16x4_f32(
            false, a, false, bb, (short)0, acc, false, false);
      }
      float* dst = ws + Qo[w] + ((size_t)b * NN + node0) * DD + ct * 16 + m16;
      #pragma unroll
      for (int r = 0; r < 8; ++r) dst[(size_t)(r + 8 * half) * DD] = acc[r];
    }
  }
}

// ---- per-(node,head) 3x3 cross-behavior attention ----
__global__ void k_attn(float* __restrict__ ws) {
  long long gid = (long long)blockIdx.x * blockDim.x + threadIdx.x;
  if (gid >= (long long)NN * NHEAD) return;
  int n = (int)(gid >> 2), h = (int)(gid & 3);
  float q[3][16], k[3][16], v[3][16];
  #pragma unroll
  for (int b = 0; b < 3; ++b) {
    size_t o = ((size_t)b * NN + n) * DD + h * 16;
    #pragma unroll
    for (int j = 0; j < 4; ++j) {
      float4 fq = *(const float4*)(ws + OFF_Q + o + j * 4);
      float4 fk = *(const float4*)(ws + OFF_K + o + j * 4);
      float4 fv = *(const float4*)(ws + OFF_V + o + j * 4);
      q[b][4*j] = fq.x; q[b][4*j+1] = fq.y; q[b][4*j+2] = fq.z; q[b][4*j+3] = fq.w;
      k[b][4*j] = fk.x; k[b][4*j+1] = fk.y; k[b][4*j+2] = fk.z; k[b][4*j+3] = fk.w;
      v[b][4*j] = fv.x; v[b][4*j+1] = fv.y; v[b][4*j+2] = fv.z; v[b][4*j+3] = fv.w;
    }
  }
  #pragma unroll
  for (int qb = 0; qb < 3; ++qb) {
    float s[3];
    #pragma unroll
    for (int kb = 0; kb < 3; ++kb) {
      float d = 0.f;
      #pragma unroll
      for (int j = 0; j < 16; ++j) d += q[qb][j] * k[kb][j];
      s[kb] = d * 0.25f;                         // 1/sqrt(16)
    }
    float m = fmaxf(s[0], fmaxf(s[1], s[2]));
    float p0 = expf(s[0] - m), p1 = expf(s[1] - m), p2 = expf(s[2] - m);
    float inv = 1.f / (p0 + p1 + p2);
    p0 *= inv; p1 *= inv; p2 *= inv;
    size_t o = ((size_t)qb * NN + n) * DD + h * 16;
    #pragma unroll
    for (int j = 0; j < 4; ++j) {
      float4 ov;
      ov.x = p0*v[0][4*j]   + p1*v[1][4*j]   + p2*v[2][4*j];
      ov.y = p0*v[0][4*j+1] + p1*v[1][4*j+1] + p2*v[2][4*j+1];
      ov.z = p0*v[0][4*j+2] + p1*v[1][4*j+2] + p2*v[2][4*j+2];
      ov.w = p0*v[0][4*j+3] + p1*v[1][4*j+3] + p2*v[2][4*j+3];
      *(float4*)(ws + OFF_ATT + o + j * 4) = ov;
    }
  }
}

// ---- att@Wo (WMMA) + residual + LayerNorm + sum over behaviors -> g ----
__global__ void __launch_bounds__(64)
k_oproj(const float* __restrict__ user_emb, const float* __restrict__ item_emb,
        const float* __restrict__ wo, const float* __restrict__ bo,
        const float* __restrict__ ln_g, const float* __restrict__ ln_b,
        float* __restrict__ ws) {
  __shared__ __align__(16) float sWp[DD * DD];
  __shared__ __align__(16) float sBo[DD], sG[DD], sBt[DD];
  __shared__ __align__(16) float sX[2][16 * 68];
  int wid = threadIdx.x >> 5, lane = threadIdx.x & 31;
  int half = lane >> 4, m16 = lane & 15;
  long long t = (long long)blockIdx.x * 2 + wid;
  if (t >= (long long)(NN / 16)) t = (long long)(NN / 16) - 1;
  int node0 = (int)t * 16;

  for (int i = threadIdx.x; i < DD * DD; i += 64) {
    int k = i >> 6, n = i & 63;
    sWp[(k >> 1) * 128 + n * 2 + (k & 1)] = wo[i];
  }
  for (int i = threadIdx.x; i < DD; i += 64) { sBo[i] = bo[i]; sG[i] = ln_g[i]; sBt[i] = ln_b[i]; }
  __syncthreads();

  float* xt = sX[wid];
  v8f gacc[4];
  #pragma unroll
  for (int ct = 0; ct < 4; ++ct) gacc[ct] = (v8f)0.f;

  for (int b = 0; b < 3; ++b) {
    // stage att tile (per-wave buffer)
    for (int idx = lane; idx < 256; idx += 32) {
      int row = idx >> 4, seg = idx & 15;
      float4 av = *(const float4*)(ws + OFF_ATT + ((size_t)b * NN + node0 + row) * DD + seg * 4);
      xt[row * 68 + seg * 4 + 0] = av.x; xt[row * 68 + seg * 4 + 1] = av.y;
      xt[row * 68 + seg * 4 + 2] = av.z; xt[row * 68 + seg * 4 + 3] = av.w;
    }
    __builtin_amdgcn_s_barrier();   // block-uniform; fences LDS for our wave

    v8f hacc[4];
    for (int ct = 0; ct < 4; ++ct) {
      float bias = sBo[ct * 16 + m16];
      v8f acc = {bias, bias, bias, bias, bias, bias, bias, bias};
      #pragma unroll
      for (int ks = 0; ks < 16; ++ks) {
        int kb = ks * 4 + 2 * half;
        v2f a = *(const v2f*)(xt + m16 * 68 + kb);
        v2f bb = *(const v2f*)(sWp + (kb >> 1) * 128 + (ct * 16 + m16) * 2);
        acc = __builtin_amdgcn_wmma_f32_16x16x4_f32(
            false, a, false, bb, (short)0, acc, false, false);
      }
      // residual: + x[node, b, col]
      #pragma unroll
      for (int r = 0; r < 8; ++r) {
        int node = node0 + r + 8 * half, col = ct * 16 + m16;
        float xv;
        if (node < NU) xv = user_emb[((size_t)node * NB + b) * DD + col];
        else {
          size_t o = ((size_t)(node - NU) * NB + b) * DD + col;
          xv = item_emb[o] + 2.f * ws[OFF_AGG + o];
        }
        acc[r] += xv;
      }
      hacc[ct] = acc;
    }
    // LayerNorm per row (a row lives in one VGPR idx across 16 lanes of a half)
    float mu8[8], rs8[8];
    #pragma unroll
    for (int r = 0; r < 8; ++r) {
      float s = hacc[0][r] + hacc[1][r] + hacc[2][r] + hacc[3][r];
      #pragma unroll
      for (int m = 8; m >= 1; m >>= 1) s += __shfl_xor(s, m, 16);
      mu8[r] = s * (1.f / 64.f);
    }
    #pragma unroll
    for (int r = 0; r < 8; ++r) {
      float vs = 0.f;
      #pragma unroll
      for (int ct = 0; ct < 4; ++ct) {
        float d = hacc[ct][r] - mu8[r];
        vs += d * d;
      }
      #pragma unroll
      for (int m = 8; m >= 1; m >>= 1) vs += __shfl_xor(vs, m, 16);
      rs8[r] = rsqrtf(vs * (1.f / 64.f) + 1e-5f);
    }
    #pragma unroll
    for (int ct = 0; ct < 4; ++ct) {
      float g = sG[ct * 16 + m16], bb = sBt[ct * 16 + m16];
      #pragma unroll
      for (int r = 0; r < 8; ++r)
        gacc[ct][r] += (hacc[ct][r] - mu8[r]) * rs8[r] * g + bb;
    }
    __builtin_amdgcn_s_barrier();   // before next b overwrites xt
  }
  #pragma unroll
  for (int ct = 0; ct < 4; ++ct)
    #pragma unroll
    for (int r = 0; r < 8; ++r)
      ws[OFF_G + (size_t)(node0 + r + 8 * half) * DD + ct * 16 + m16] = gacc[ct][r];
}

// ---- gather batch rows ----
__global__ void k_gather(const int* __restrict__ users, const int* __restrict__ pos,
                         const int* __restrict__ neg, const float* __restrict__ ws,
                         float* __restrict__ out) {
  int gid = blockIdx.x * blockDim.x + threadIdx.x;
  if (gid >= 3 * NBATCH * 16) return;
  int row = gid >> 4, seg = gid & 15, n;
  if (row < NBATCH) n = users[row];
  else if (row < 2 * NBATCH) n = NU + pos[row - NBATCH];
  else n = NU + neg[row - 2 * NBATCH];
  *(float4*)(out + (size_t)row * DD + seg * 4) =
      *(const float4*)(ws + OFF_G + (size_t)n * DD + seg * 4);
}

extern "C" void kernel_launch(void* const* d_in, const int* in_sizes, int n_in,
                              void* d_out, int out_size, void* d_ws, size_t ws_size,
                              hipStream_t stream) {
  const float* user_emb = (const float*)d_in[0];
  const float* item_emb = (const float*)d_in[1];
  const float* alpha_w  = (const float*)d_in[2];
  const float* alpha_b  = (const float*)d_in[3];
  const float* wq = (const float*)d_in[4];  const float* bq = (const float*)d_in[5];
  const float* wk = (const float*)d_in[6];  const float* bk = (const float*)d_in[7];
  const float* wv = (const float*)d_in[8];  const float* bv = (const float*)d_in[9];
  const float* wo = (const float*)d_in[10]; const float* bo = (const float*)d_in[11];
  const float* lng = (const float*)d_in[12]; const float* lnb = (const float*)d_in[13];
  const int* ei    = (const int*)d_in[14];
  const int* users = (const int*)d_in[15];
  const int* pos   = (const int*)d_in[16];
  const int* neg   = (const int*)d_in[17];
  float* out = (float*)d_out;
  float* ws  = (float*)d_ws;

  k_zero<<<(10200000u + 255) / 256, 256, 0, stream>>>(ws, 10200000u);
  k_edge_phase<<<375000, 256, 0, stream>>>(user_emb, item_emb, alpha_w, alpha_b, ei, ws, out);
  k_edge_exp<<<(3000000 + 255) / 256, 256, 0, stream>>>(ei, ws, out);
  k_norm_scatter<<<375000, 256, 0, stream>>>(user_emb, ei, ws, out);
  k_qkv<<<(28125 + 1) / 2, 64, 0, stream>>>(user_emb, item_emb, wq, bq, wk, bk, wv, bv, ws);
  k_attn<<<(600000 + 255) / 256, 256, 0, stream>>>(ws);
  k_oproj<<<(9375 + 1) / 2, 64, 0, stream>>>(user_emb, item_emb, wo, bo, lng, lnb, ws);
  k_gather<<<(3 * NBATCH * 16 + 255) / 256, 256, 0, stream>>>(users, pos, neg, ws, out);
}